// ADAM_SINDy_MODEL_64476049048178
// MI455X (gfx1250) — compile-verified
//
#include <hip/hip_runtime.h>
#include <hip/hip_bf16.h>

typedef __attribute__((ext_vector_type(2))) float v2f;
typedef __attribute__((ext_vector_type(8))) float v8f;

#define NXC 21
#define NDC 5
#define NTERMS 568
#define KPAD 28        // 27-dim p vector padded to 28
#define NPAD 48        // 42 G-rows padded to 48
#define GSZ (NPAD * KPAD)   // 1344 floats
// workspace layout: [0 .. GSZ) = G matrix (row n, col k), [GSZ] = c0

// ---------------- prep kernel: build masked coefficient matrix G ----------------
__global__ void sindy_prep_kernel(const float* __restrict__ a,
                                  const int* __restrict__ drug_idx,
                                  const int* __restrict__ bilin_idx,
                                  const int* __restrict__ mm2_idx,
                                  const unsigned char* __restrict__ uses_self,
                                  float* __restrict__ Gws) {
    int tid = threadIdx.x;
    for (int i = tid; i < GSZ + 1; i += 256) Gws[i] = 0.0f;
    __syncthreads();
    for (int k = tid; k < NTERMS; k += 256) {
        float av = a[k];
        bool zero = (uses_self[k] ? (av > 0.0f) : (av < 0.0f)) && (k >= 2);
        float c = zero ? 0.0f : av;
        if (k == 0) {
            Gws[GSZ] = c;                                    // c0 * con handled in epilogue
        } else if (k < 22) {                                 // linear terms -> u const part
            Gws[(k - 1) * KPAD + 0] = c;
        } else if (k < 127) {                                // drug terms x_i * drug_d
            int p = k - 22;
            int i = drug_idx[2 * p], d = drug_idx[2 * p + 1];
            Gws[i * KPAD + 1 + NXC + d] = c;
        } else if (k < 337) {                                // bilinear x_i * x_j (i<j)
            int p = k - 127;
            int i = bilin_idx[2 * p], j = bilin_idx[2 * p + 1];
            Gws[i * KPAD + 1 + j] = c;
        } else if (k < 547) {                                // mm2: r_i * x_j (i<j)
            int p = k - 337;
            int i = mm2_idx[2 * p], j = mm2_idx[2 * p + 1];
            Gws[(NXC + i) * KPAD + 1 + j] = c;
        } else {                                             // hill: r_i -> v const part
            int i = k - 547;
            Gws[(NXC + i) * KPAD + 0] = c;
        }
    }
}

// ---------------- main kernel: WMMA GEMM + nonlinear epilogue ----------------
// block = 256 threads = 8 waves; each wave owns a 16-element tile -> 128 elems/block
__global__ void __launch_bounds__(256)
sindy_main_kernel(const float* __restrict__ cand,
                  const float* __restrict__ Gws,
                  float* __restrict__ out) {
    // LDS: G (1344) + per-wave { P[16][28]=448, R[16][21]=336, con[16] } * 8
    __shared__ float smem[GSZ + 8 * (16 * KPAD + 16 * NXC + 16)];
    float* Gl = smem;
    int wav = threadIdx.x >> 5;
    int lane = threadIdx.x & 31;
    int hi = lane >> 4;         // 0: lanes 0-15, 1: lanes 16-31
    int lm = lane & 15;

    float* Pl  = smem + GSZ + wav * (16 * KPAD + 16 * NXC + 16);
    float* Rl  = Pl + 16 * KPAD;
    float* Cnl = Rl + 16 * NXC;

    int e0 = blockIdx.x * 128 + wav * 16;   // first element of this wave's tile

    // stage G (all 256 threads)
    for (int i = threadIdx.x; i < GSZ; i += 256) Gl[i] = Gws[i];

    // stage this wave's 16 rows of candidates, coalesced
    const float* src = cand + (size_t)e0 * 27;
    for (int idx = lane; idx < 16 * 27; idx += 32) {
        int e = idx / 27, c = idx % 27;
        Pl[e * KPAD + c] = src[idx];
    }
    // fixups: save con, set p[0]=1, p[27]=0  (same-wave LDS, in-order via DScnt)
    if (lane < 16) {
        Cnl[lane] = Pl[lane * KPAD + 0];
        Pl[lane * KPAD + 0] = 1.0f;
        Pl[lane * KPAD + 27] = 0.0f;
    }
    // r_i = x_i / (0.5 + x_i)  (this also equals the hill ratio since K2==K3)
    for (int idx = lane; idx < 16 * NXC; idx += 32) {
        int e = idx / NXC, j = idx % NXC;
        float v = Pl[e * KPAD + 1 + j];
        Rl[idx] = v / (0.5f + v);
    }
    __syncthreads();

    // GEMM: Y(16x48) = P(16x28) @ G^T(28x48) via 7 K-chunks x 3 N-tiles of WMMA f32 16x16x4
    v8f acc0 = {}, acc1 = {}, acc2 = {};
#pragma unroll
    for (int kc = 0; kc < 7; ++kc) {
        int k0 = 4 * kc + 2 * hi;            // ISA A/B layout: upper lanes carry K+2
        v2f A;
        A.x = Pl[lm * KPAD + k0];
        A.y = Pl[lm * KPAD + k0 + 1];
        v2f B0, B1, B2;
        B0.x = Gl[(lm +  0) * KPAD + k0];  B0.y = Gl[(lm +  0) * KPAD + k0 + 1];
        B1.x = Gl[(lm + 16) * KPAD + k0];  B1.y = Gl[(lm + 16) * KPAD + k0 + 1];
        B2.x = Gl[(lm + 32) * KPAD + k0];  B2.y = Gl[(lm + 32) * KPAD + k0 + 1];
        acc0 = __builtin_amdgcn_wmma_f32_16x16x4_f32(false, A, false, B0, (short)0, acc0, false, false);
        acc1 = __builtin_amdgcn_wmma_f32_16x16x4_f32(false, A, false, B1, (short)0, acc1, false, false);
        acc2 = __builtin_amdgcn_wmma_f32_16x16x4_f32(false, A, false, B2, (short)0, acc2, false, false);
    }

    float c0 = Gws[GSZ];

    // epilogue: out[m] = c0*con + sum_n x[m][n]*Y[m][n<21] + sum r[m][n-21]*Y[m][21<=n<42]
    // C/D layout: VGPR k holds row m = k + 8*hi, column n = 16*t + lm
#pragma unroll
    for (int k = 0; k < 8; ++k) {
        int m = k + 8 * hi;
        // tile 0: n = lm in [0,16) -> weight x[m][n]
        float s = acc0[k] * Pl[m * KPAD + 1 + lm];
        // tile 1: n = 16+lm -> x for n<21, else r[n-21]
        int n1 = 16 + lm;
        float w1 = (n1 < NXC) ? Pl[m * KPAD + 1 + n1] : Rl[m * NXC + (n1 - NXC)];
        s += acc1[k] * w1;
        // tile 2: n = 32+lm -> r[n-21] for n<42, else padding
        int n2 = 32 + lm;
        float w2 = (n2 < 2 * NXC) ? Rl[m * NXC + (n2 - NXC)] : 0.0f;
        s += acc2[k] * w2;
        // reduce the 16 columns within each half-wave
        s += __shfl_xor(s, 1, 32);
        s += __shfl_xor(s, 2, 32);
        s += __shfl_xor(s, 4, 32);
        s += __shfl_xor(s, 8, 32);
        if (lm == 0) out[e0 + m] = s + c0 * Cnl[m];
    }
}

extern "C" void kernel_launch(void* const* d_in, const int* in_sizes, int n_in,
                              void* d_out, int out_size, void* d_ws, size_t ws_size,
                              hipStream_t stream) {
    const float* cand            = (const float*)d_in[0];
    const float* a               = (const float*)d_in[1];
    const int* drug_idx          = (const int*)d_in[3];
    const int* bilin_idx         = (const int*)d_in[4];
    const int* mm2_idx           = (const int*)d_in[5];
    const unsigned char* uses_sf = (const unsigned char*)d_in[7];
    float* Gws = (float*)d_ws;
    float* out = (float*)d_out;

    sindy_prep_kernel<<<1, 256, 0, stream>>>(a, drug_idx, bilin_idx, mm2_idx, uses_sf, Gws);
    // 262144 elements / 128 per block = 2048 blocks
    int nblocks = out_size / 128;
    sindy_main_kernel<<<nblocks, 256, 0, stream>>>(cand, Gws, out);
}